// QLayer_65481071402449
// MI455X (gfx1250) — compile-verified
//
#include <hip/hip_runtime.h>
#include <math.h>

// ============================================================================
// QLayer (14-qubit RX -> Rot -> CNOT-ladder -> <Z_w>) — analytic closed form.
//
// Derivation:
//   * RX layer + Rot layer keep the state a tensor product of 1-qubit states.
//   * The CNOT ladder permutes basis states: bit w -> XOR_{j<=w} z_j, hence
//       <Z_w>_after = < prod_{j<=w} Z_j >_before = prod_{j<=w} t_j
//     with per-qubit  t_j = cos(theta_j)*cos(x_j) - sin(theta_j)*sin(phi_j)*sin(x_j)
//     (omega_j drops out of the Z expectation).
//   * out[b,w] = (prefix_prod + 1) / 2.
//
// Prefix product via the matrix pipe: prefix scan == matmul with a lower-
// triangular ones matrix. We do it in log2 space (fp32) with chained
// V_WMMA_F32_16X16X4_F32 (4 chunks of K=4 -> K=16), plus a second WMMA chain
// that prefix-counts sign bits (exact small-integer fp32 sums) for the parity.
//
// One wave per 16-batch-row tile; 32 tiles cover BATCH=512. Qubit columns
// 14,15 padded with t=1 (log2=0, sign=0) so they contribute nothing.
// ============================================================================

typedef __attribute__((ext_vector_type(2))) float v2f;
typedef __attribute__((ext_vector_type(8))) float v8f;

#define NQ    14
#define BATCH 512

__global__ __launch_bounds__(32) void qlayer_wmma_kernel(
    const float* __restrict__ x,       // (512, 14)
    const float* __restrict__ params,  // (14, 3)
    float* __restrict__ out)           // (512, 14)
{
  __shared__ float Am[16][16];  // log2|t|   (0 in padded cols)
  __shared__ float As[16][16];  // 1.0 if t<0 else 0.0

  const int lane = threadIdx.x;      // 0..31 (wave32)
  const int b0   = blockIdx.x * 16;  // batch tile base row

  // ---- Stage 1: elementwise t = cos(th)cos(x) - sin(th)sin(phi)sin(x) ------
  // 16x16 entries, 8 per lane; uniform loop (no EXEC divergence at the WMMAs).
  for (int e = lane; e < 256; e += 32) {
    const int m = e >> 4;   // batch row within tile
    const int j = e & 15;   // qubit (14,15 = identity padding)
    float mlog = 0.0f, sgn = 0.0f;
    if (j < NQ) {
      const float phi   = params[j * 3 + 0];
      const float theta = params[j * 3 + 1];
      const float xv    = x[(b0 + m) * NQ + j];
      const float t = cosf(theta) * cosf(xv)
                    - sinf(theta) * sinf(phi) * sinf(xv);
      // clamp magnitude away from 0 so log2 stays finite (Inf*0 in the WMMA
      // would make NaN); exp2 of the clamped log underflows to 0 == product 0.
      mlog = log2f(fmaxf(fabsf(t), 1e-30f));
      sgn  = (t < 0.0f) ? 1.0f : 0.0f;
    }
    Am[m][j] = mlog;
    As[m][j] = sgn;
  }
  __syncthreads();

  // ---- Stage 2: prefix scan = triangular matmul on the matrix pipe ---------
  // A layout (32-bit 16x4, documented): lane&15 = M; VGPR v, lane-half h -> K = 4c + 2h + v.
  // B layout (4x16, C/D-style row pairing): lane&15 = N; VGPR v, half h -> K = 4c + 2h + v.
  // D layout (documented): lane&15 = N; VGPR r, half h -> M = r + 8h.
  const int ln = lane & 15;  // A: M row | B/D: N column (wire)
  const int h  = lane >> 4;

  v8f dM = {0.f, 0.f, 0.f, 0.f, 0.f, 0.f, 0.f, 0.f};  // prefix-sum log2|t|
  v8f dS = {0.f, 0.f, 0.f, 0.f, 0.f, 0.f, 0.f, 0.f};  // prefix-count of negatives

#pragma unroll
  for (int c = 0; c < 4; ++c) {
    const int k0 = 4 * c + 2 * h;
    v2f aM, aS, bL;
    aM[0] = Am[ln][k0];     aM[1] = Am[ln][k0 + 1];
    aS[0] = As[ln][k0];     aS[1] = As[ln][k0 + 1];
    bL[0] = (k0     <= ln) ? 1.0f : 0.0f;   // L[k][n] = (k <= n)
    bL[1] = (k0 + 1 <= ln) ? 1.0f : 0.0f;
    // 8 args: (neg_a, A, neg_b, B, c_mod, C, reuse_a, reuse_b)
    dM = __builtin_amdgcn_wmma_f32_16x16x4_f32(false, aM, false, bL,
                                               (short)0, dM, false, false);
    dS = __builtin_amdgcn_wmma_f32_16x16x4_f32(false, aS, false, bL,
                                               (short)0, dS, false, false);
  }

  // ---- Stage 3: reconstruct products and write (prod + 1)/2 ----------------
  const int w = ln;  // wire index = N column
  if (w < NQ) {
#pragma unroll
    for (int r = 0; r < 8; ++r) {
      const int m = r + 8 * h;                 // batch row within tile
      float p = exp2f(dM[r]);                  // |prefix product|
      const int negs = (int)(dS[r] + 0.25f);   // exact small integer
      if (negs & 1) p = -p;
      out[(b0 + m) * NQ + w] = 0.5f * (p + 1.0f);
    }
  }
}

extern "C" void kernel_launch(void* const* d_in, const int* in_sizes, int n_in,
                              void* d_out, int out_size, void* d_ws, size_t ws_size,
                              hipStream_t stream) {
  const float* x      = (const float*)d_in[0];  // (512, 14) fp32
  const float* params = (const float*)d_in[1];  // (14, 3)  fp32
  float* out          = (float*)d_out;          // (512, 14) fp32

  // 32 tiles of 16 batch rows; one wave32 per block.
  qlayer_wmma_kernel<<<BATCH / 16, 32, 0, stream>>>(x, params, out);
}